// _RoiAlign_83743272337919
// MI455X (gfx1250) — compile-verified
//
#include <hip/hip_runtime.h>
#include <math.h>
#include <stdint.h>

// ROI-Align over 5-level FPN, fp32, NHWC. Memory-bound gather (≈0.4 FLOP/byte):
// optimized for the CDNA5 data-movement path (ASYNCcnt copies), not WMMA.
//  - 3D grid (14 rows, 128 boxes, B images); 256 threads = 8 wave32 per block
//  - stage 14 px × 4 corners × 1KB = 56KB into LDS via global_load_async_to_lds_b128
//  - s_wait_asynccnt 0 + barrier, then 4-tap blend from LDS, NT b128 stores.

#define CROP 14
#define NCH  256
#define NQ   (NCH / 4)   // 64 channel-quads of float4

typedef float v4f __attribute__((ext_vector_type(4)));

__device__ __forceinline__ void async_ld_b128(unsigned lds_byte_off, const void* gaddr) {
  // VDST = LDS byte address (VGPR), VADDR = 64-bit global address (VGPR pair)
  asm volatile("global_load_async_to_lds_b128 %0, %1, off"
               :: "v"(lds_byte_off), "v"(gaddr)
               : "memory");
}

__device__ __forceinline__ void wait_async0() {
  asm volatile("s_wait_asynccnt 0" ::: "memory");
}

__global__ __launch_bounds__(256)
void roi_align_fpn(const float* __restrict__ img_shape,
                   const float* __restrict__ boxes,
                   const float* __restrict__ f0,
                   const float* __restrict__ f1,
                   const float* __restrict__ f2,
                   const float* __restrict__ f3,
                   const float* __restrict__ f4,
                   float* __restrict__ out,
                   int N)
{
  __shared__ __align__(16) float smem[CROP * 4 * NCH];   // 56 KB

  const int tid = threadIdx.x;
  const int y   = blockIdx.x;                 // output row   0..13
  const int n   = blockIdx.y;                 // box          0..N-1
  const int b   = blockIdx.z;                 // image        0..B-1
  const int bn  = b * N + n;

  const float H = img_shape[1];
  const float W = img_shape[2];

  const float* bp = boxes + (size_t)bn * 4;
  const float bx1 = bp[0], by1 = bp[1], bx2 = bp[2], by2 = bp[3];

  // ---- level mapping: clip(floor(1 + log2(sqrt(w*h)/224 + 1e-7)), 0, 4) ----
  const float bwd = bx2 - bx1;
  const float bhd = by2 - by1;
  const float sz  = sqrtf(bwd * bhd);
  float lvlf = floorf(1.0f + log2f(sz / 224.0f + 1e-7f));
  lvlf = fminf(fmaxf(lvlf, 0.0f), 4.0f);
  const int lvl = (int)lvlf;

  const float* feat;
  int fdim;
  if      (lvl == 0) { feat = f0; fdim = 128; }
  else if (lvl == 1) { feat = f1; fdim = 64;  }
  else if (lvl == 2) { feat = f2; fdim = 32;  }
  else if (lvl == 3) { feat = f3; fdim = 16;  }
  else               { feat = f4; fdim = 8;   }
  feat += (size_t)b * fdim * fdim * NCH;

  // ---- normalized box for this level (matches reference formulas) ----
  const float fdimf = (float)fdim;
  const float den   = fdimf - 1.0f;
  const float y1n = by1 / H * fdimf / den;
  const float x1n = bx1 / W * fdimf / den;
  const float y2n = (by2 / H * fdimf - 1.0f) / den;
  const float x2n = (bx2 / W * fdimf - 1.0f) / den;

  // ---- y sampling for this output row ----
  const float ty   = (float)y / 13.0f;
  const float in_y = (y1n + ty * (y2n - y1n)) * den;
  const bool  vy   = (in_y >= 0.0f) && (in_y <= den);
  const float y0f  = floorf(in_y);
  const float wy   = in_y - y0f;
  const int y0i = (int)fminf(fmaxf(y0f,        0.0f), den);
  const int y1i = (int)fminf(fmaxf(y0f + 1.0f, 0.0f), den);

  // ---- Phase 1: async-stage all 14 px × 4 corners × 1KB into LDS ----
  // thread -> (corner ct in 0..3, channel-quad cq in 0..63); loop over pixels.
  const unsigned lbase = (unsigned)(unsigned long long)(uintptr_t)&smem[0];
  const int ct = tid >> 6;
  const int cq = tid & (NQ - 1);
  const int yc = (ct < 2) ? y0i : y1i;
  const float* rowbase = feat + (size_t)yc * fdim * NCH + cq * 4;

  for (int p = 0; p < CROP; ++p) {
    const float tx   = (float)p / 13.0f;
    const float in_x = (x1n + tx * (x2n - x1n)) * den;
    const float x0f  = floorf(in_x);
    const int x0i = (int)fminf(fmaxf(x0f,        0.0f), den);
    const int x1i = (int)fminf(fmaxf(x0f + 1.0f, 0.0f), den);
    const int xc  = (ct & 1) ? x1i : x0i;
    const unsigned loff = lbase + (unsigned)((((p * 4 + ct) * NCH) + cq * 4) * 4);
    async_ld_b128(loff, rowbase + (size_t)xc * NCH);
  }
  wait_async0();        // per-wave: all async copies into LDS complete
  __syncthreads();      // cross-wave visibility within workgroup

  // ---- Phase 2: 4-tap bilinear blend from LDS, NT stores ----
  const v4f* sv = reinterpret_cast<const v4f*>(smem);
  const size_t opix_base = ((size_t)bn * CROP + y) * CROP;  // in output pixels

  for (int it = 0; it < 4; ++it) {
    const int i = it * 256 + tid;            // 0..1023, valid < 896
    if (i < CROP * NQ) {
      const int p = i >> 6;                  // output column
      const int q = i & (NQ - 1);            // channel quad

      const float tx   = (float)p / 13.0f;
      const float in_x = (x1n + tx * (x2n - x1n)) * den;
      const bool  vx   = (in_x >= 0.0f) && (in_x <= den);
      const float x0f  = floorf(in_x);
      const float wx   = in_x - x0f;
      const bool  keep = vy && vx;

      const v4f v00 = sv[(p * 4 + 0) * NQ + q];
      const v4f v01 = sv[(p * 4 + 1) * NQ + q];
      const v4f v10 = sv[(p * 4 + 2) * NQ + q];
      const v4f v11 = sv[(p * 4 + 3) * NQ + q];

      v4f r;
      {
        const v4f top = v00 + (v01 - v00) * wx;
        const v4f bot = v10 + (v11 - v10) * wx;
        r = top + (bot - top) * wy;
      }
      if (!keep) r = (v4f){0.0f, 0.0f, 0.0f, 0.0f};

      float* optr = out + (opix_base + p) * NCH + q * 4;
      __builtin_nontemporal_store(r, (v4f*)optr);   // keep FPN L2-resident
    }
  }
}

extern "C" void kernel_launch(void* const* d_in, const int* in_sizes, int n_in,
                              void* d_out, int out_size, void* d_ws, size_t ws_size,
                              hipStream_t stream) {
  (void)n_in; (void)d_ws; (void)ws_size; (void)out_size;

  const float* img   = (const float*)d_in[0];   // [B, H, W, 3] as 4 floats
  const float* boxes = (const float*)d_in[1];   // (B, N, 4)
  // d_in[2] = scores: unused by the reference
  const float* f0 = (const float*)d_in[3];
  const float* f1 = (const float*)d_in[4];
  const float* f2 = (const float*)d_in[5];
  const float* f3 = (const float*)d_in[6];
  const float* f4 = (const float*)d_in[7];

  const int N = 128;
  const int B = in_sizes[1] / (N * 4);

  dim3 grid((unsigned)CROP, (unsigned)N, (unsigned)B);
  roi_align_fpn<<<grid, 256, 0, stream>>>(img, boxes, f0, f1, f2, f3, f4,
                                          (float*)d_out, N);
}